// MyGIN_52948356825765
// MI455X (gfx1250) — compile-verified
//
#include <hip/hip_runtime.h>

#define Nn 50000
#define Ee 800000
#define Dd 64
#define Ll 5
#define Gg 500
#define Cc 10

typedef float v2f __attribute__((ext_vector_type(2)));
typedef float v8f __attribute__((ext_vector_type(8)));
typedef unsigned int v4u __attribute__((ext_vector_type(4)));
typedef int v4i __attribute__((ext_vector_type(4)));
typedef int v8i __attribute__((ext_vector_type(8)));

// ---------------------------------------------------------------- utilities
__global__ __launch_bounds__(256) void fill_zero(float* __restrict__ p, int n) {
    int i = blockIdx.x * 256 + threadIdx.x;
    if (i < n) p[i] = 0.0f;
}

__global__ __launch_bounds__(256) void copy_f32(const float* __restrict__ s,
                                                float* __restrict__ d, int n) {
    int i = blockIdx.x * 256 + threadIdx.x;
    if (i < n) d[i] = s[i];
}

// ------------------------------------------------- edge scatter (segment_sum)
// 16 threads per edge, float4 gather + 4 f32 atomics. h is L2-resident (12.8MB).
__global__ __launch_bounds__(256) void edge_scatter(const float* __restrict__ h,
                                                    const long long* __restrict__ ei,
                                                    float* __restrict__ agg) {
    int tid = blockIdx.x * 256 + threadIdx.x;       // E*16 threads
    int e = tid >> 4;
    if (e >= Ee) return;
    int f = (tid & 15) * 4;
    long long s = ei[e];
    long long d = ei[Ee + e];
    const float4 v = *(const float4*)(h + s * Dd + f);
    float* o = agg + d * Dd + f;
    atomicAdd(o + 0, v.x);
    atomicAdd(o + 1, v.y);
    atomicAdd(o + 2, v.z);
    atomicAdd(o + 3, v.w);
}

// ------------------------------------------------------- TDM weight staging
// One 2D tensor_load_to_lds of a 64x64 f32 tile, LDS row stride padded to 72
// dwords (pad_interval=64 DW -> code 5, pad_amount=8 DW -> code 7) so the
// B-operand reads (rows k0..k0+3 split across half-waves) are bank-conflict
// free (2*72 mod 64 == 16).
static __device__ __forceinline__ void tdm_load_w64x64(const float* __restrict__ gsrc,
                                                       unsigned lds_off) {
    unsigned long long ga = (unsigned long long)(size_t)gsrc;
    v4u g0;
    g0.x = 1u;                                            // count=1 (valid), user mode
    g0.y = lds_off;                                       // lds_addr (bytes)
    g0.z = (unsigned)(ga & 0xffffffffull);                // global_addr[31:0]
    g0.w = (unsigned)((ga >> 32) & 0x1ffffffull) | (2u << 30);  // addr[56:32] | type=2
    v8i g1;
    g1[0] = (2 << 16) | (1 << 20) | (5 << 22) | (7 << 25);
            // wg_mask=0 | data_size=4B | pad_enable | pad_interval=64DW | pad_amount=8DW
    g1[1] = 64 << 16;   // tensor_dim0 = 64 (bits 79:48, low half)
    g1[2] = 64 << 16;   // tensor_dim0 hi=0 | tensor_dim1 = 64 (low half)
    g1[3] = 64 << 16;   // tensor_dim1 hi=0 | tile_dim0 = 64
    g1[4] = 64;         // tile_dim1 = 64 | tile_dim2 = 0
    g1[5] = 64;         // tensor_dim0_stride = 64 (elements)
    g1[6] = 0;
    g1[7] = 0;
    v4i z4 = {0, 0, 0, 0};
    v8i z8 = {0, 0, 0, 0, 0, 0, 0, 0};
    // amdgpu-toolchain (clang-23) 6-arg form: groups 2/3/4 disabled (2D tensor)
    __builtin_amdgcn_tensor_load_to_lds(g0, g1, z4, z4, z8, 0);
}

// --------------------------------------------- fused GIN MLP (2 GEMMs, WMMA)
// Block = 256 threads (8 waves), 64-row tile, D=K=64.
// Wave w owns rows [16*(w/2),+16), cols [32*(w%2),+32) => two 16x16 WMMA tiles.
// Weights arrive via TDM (async tensor DMA) double-buffered in LDS while the
// waves build the (1+eps)h + agg input tile.
__global__ __launch_bounds__(256) void gin_mlp(const float* __restrict__ h,
                                               const float* __restrict__ agg,
                                               const float* __restrict__ eps_arr, int layer,
                                               const float* __restrict__ W1g,
                                               const float* __restrict__ b1g,
                                               const float* __restrict__ W2g,
                                               const float* __restrict__ b2g,
                                               float* __restrict__ z_out,
                                               float* __restrict__ colsum,
                                               float* __restrict__ colsumsq) {
    __shared__ float Xs[64][68];    // stride 68: rows 16B aligned, A-reads conflict-free
    __shared__ float Zs[64][68];
    __shared__ float Ws1[64][72];   // stride 72 via TDM padding: B-reads conflict-free
    __shared__ float Ws2[64][72];

    const int tid  = threadIdx.x;
    const int wave = tid >> 5;
    const int lane = tid & 31;
    const int hi   = lane >> 4;     // lane-half selects upper K pair / M+8
    const int l15  = lane & 15;
    const int rb   = (wave >> 1) * 16;
    const int cb   = (wave & 1) * 32;
    const int row0 = blockIdx.x * 64;
    const float epl = 1.0f + eps_arr[layer];

    // wave 0: kick off both weight DMAs (in-order on TENSORcnt)
    if (tid < 32) {
        tdm_load_w64x64(W1g + layer * 4096, (unsigned)(size_t)&Ws1[0][0]);
        tdm_load_w64x64(W2g + layer * 4096, (unsigned)(size_t)&Ws2[0][0]);
    }

    // all waves: stage input tile ((1+eps)h + agg), float4 vectorized
    for (int i = tid; i < 1024; i += 256) {
        int r = i >> 4;
        int c = (i & 15) * 4;
        int gr = row0 + r;
        float4 v = {0.0f, 0.0f, 0.0f, 0.0f};
        if (gr < Nn) {
            const float4 hv = *(const float4*)(h + gr * 64 + c);
            const float4 av = *(const float4*)(agg + gr * 64 + c);
            v.x = epl * hv.x + av.x;
            v.y = epl * hv.y + av.y;
            v.z = epl * hv.z + av.z;
            v.w = epl * hv.w + av.w;
        }
        *(float4*)&Xs[r][c] = v;
    }
    if (tid < 32) __builtin_amdgcn_s_wait_tensorcnt(0);
    __syncthreads();

    // ---- GEMM1: Zs = relu(Xs @ W1 + b1)
    v8f c0 = {};
    v8f c1 = {};
#pragma unroll
    for (int k0 = 0; k0 < 64; k0 += 4) {
        v2f a = *(const v2f*)&Xs[rb + l15][k0 + 2 * hi];   // ds_load_b64
        v2f bA, bB;
        bA.x = Ws1[k0 + 2 * hi][cb + l15];
        bA.y = Ws1[k0 + 2 * hi + 1][cb + l15];
        bB.x = Ws1[k0 + 2 * hi][cb + 16 + l15];
        bB.y = Ws1[k0 + 2 * hi + 1][cb + 16 + l15];
        c0 = __builtin_amdgcn_wmma_f32_16x16x4_f32(false, a, false, bA, (short)0, c0, false, false);
        c1 = __builtin_amdgcn_wmma_f32_16x16x4_f32(false, a, false, bB, (short)0, c1, false, false);
    }
    {
        float bb0 = b1g[layer * 64 + cb + l15];
        float bb1 = b1g[layer * 64 + cb + 16 + l15];
#pragma unroll
        for (int i = 0; i < 8; ++i) {
            int r = rb + i + 8 * hi;
            Zs[r][cb + l15]      = fmaxf(c0[i] + bb0, 0.0f);
            Zs[r][cb + 16 + l15] = fmaxf(c1[i] + bb1, 0.0f);
        }
    }
    __syncthreads();

    // ---- GEMM2: z = relu(Zs @ W2 + b2)
    v8f d0 = {};
    v8f d1 = {};
#pragma unroll
    for (int k0 = 0; k0 < 64; k0 += 4) {
        v2f a = *(const v2f*)&Zs[rb + l15][k0 + 2 * hi];
        v2f bA, bB;
        bA.x = Ws2[k0 + 2 * hi][cb + l15];
        bA.y = Ws2[k0 + 2 * hi + 1][cb + l15];
        bB.x = Ws2[k0 + 2 * hi][cb + 16 + l15];
        bB.y = Ws2[k0 + 2 * hi + 1][cb + 16 + l15];
        d0 = __builtin_amdgcn_wmma_f32_16x16x4_f32(false, a, false, bA, (short)0, d0, false, false);
        d1 = __builtin_amdgcn_wmma_f32_16x16x4_f32(false, a, false, bB, (short)0, d1, false, false);
    }

    // epilogue: bias+relu, store, and BN column partial sums straight from regs
    float bb0 = b2g[layer * 64 + cb + l15];
    float bb1 = b2g[layer * 64 + cb + 16 + l15];
    float s0 = 0.f, q0 = 0.f, s1 = 0.f, q1 = 0.f;
#pragma unroll
    for (int i = 0; i < 8; ++i) {
        int r = rb + i + 8 * hi;
        int gr = row0 + r;
        float v0 = fmaxf(d0[i] + bb0, 0.0f);
        float v1 = fmaxf(d1[i] + bb1, 0.0f);
        if (gr < Nn) {
            z_out[gr * 64 + cb + l15]      = v0;
            z_out[gr * 64 + cb + 16 + l15] = v1;
            s0 += v0; q0 += v0 * v0;
            s1 += v1; q1 += v1 * v1;
        }
    }
    atomicAdd(&colsum[cb + l15], s0);
    atomicAdd(&colsumsq[cb + l15], q0);
    atomicAdd(&colsum[cb + 16 + l15], s1);
    atomicAdd(&colsumsq[cb + 16 + l15], q1);
}

// ---------------------------------------------------------------- batchnorm
__global__ __launch_bounds__(64) void bn_stats(const float* __restrict__ colsum,
                                               const float* __restrict__ colsumsq,
                                               float* __restrict__ mu,
                                               float* __restrict__ rsig) {
    int d = threadIdx.x;
    float invN = 1.0f / (float)Nn;
    float m = colsum[d] * invN;
    float v = colsumsq[d] * invN - m * m;   // biased variance
    mu[d] = m;
    rsig[d] = rsqrtf(v + 1e-5f);
}

__global__ __launch_bounds__(256) void bn_apply(const float* __restrict__ z,
                                                const float* __restrict__ mu,
                                                const float* __restrict__ rsig,
                                                const float* __restrict__ gamma,
                                                const float* __restrict__ beta, int layer,
                                                float* __restrict__ h,
                                                float* __restrict__ hcat) {
    int i = blockIdx.x * 256 + threadIdx.x;
    if (i >= Nn * 64) return;
    int n = i >> 6, d = i & 63;
    float v = (z[i] - mu[d]) * rsig[d] * gamma[layer * 64 + d] + beta[layer * 64 + d];
    h[i] = v;                                   // next-layer input
    hcat[n * (Ll * 64) + layer * 64 + d] = v;   // JumpingKnowledge cat slice
}

// ---------------------------------------------------------------- pooling
__global__ __launch_bounds__(256) void count_nodes(const long long* __restrict__ batch,
                                                   float* __restrict__ cnt) {
    int n = blockIdx.x * 256 + threadIdx.x;
    if (n < Nn) atomicAdd(&cnt[batch[n]], 1.0f);
}

__global__ __launch_bounds__(256) void pool_sum(const float* __restrict__ hcat,
                                                const long long* __restrict__ batch,
                                                float* __restrict__ pooled) {
    int tid = blockIdx.x * 256 + threadIdx.x;   // N*80 threads, float4 each
    if (tid >= Nn * 80) return;
    int n = tid / 80;
    int d = (tid % 80) * 4;
    long long g = batch[n];
    const float4 v = *(const float4*)(hcat + (size_t)n * 320 + d);
    float* o = pooled + (size_t)g * 320 + d;
    atomicAdd(o + 0, v.x);
    atomicAdd(o + 1, v.y);
    atomicAdd(o + 2, v.z);
    atomicAdd(o + 3, v.w);
}

// ------------------------------------------------------------- MLP head
__global__ __launch_bounds__(64) void head(const float* __restrict__ pooled,
                                           const float* __restrict__ cnt,
                                           const float* __restrict__ l1W,
                                           const float* __restrict__ l1b,
                                           const float* __restrict__ l2W,
                                           const float* __restrict__ l2b,
                                           float* __restrict__ out) {
    __shared__ float o1[64];
    __shared__ float logits[Cc];
    __shared__ float lse;
    int g = blockIdx.x, j = threadIdx.x;
    float inv = 1.0f / fmaxf(cnt[g], 1.0f);
    const float* p = pooled + (size_t)g * 320;
    float acc = l1b[j];
    for (int k = 0; k < 320; ++k) acc += (p[k] * inv) * l1W[k * 64 + j];
    o1[j] = fmaxf(acc, 0.0f);
    __syncthreads();
    if (j < Cc) {
        float a = l2b[j];
#pragma unroll
        for (int t = 0; t < 64; ++t) a += o1[t] * l2W[t * Cc + j];
        logits[j] = a;
    }
    __syncthreads();
    if (j == 0) {
        float m = logits[0];
        for (int c = 1; c < Cc; ++c) m = fmaxf(m, logits[c]);
        float s = 0.0f;
        for (int c = 0; c < Cc; ++c) s += expf(logits[c] - m);
        lse = m + logf(s);
    }
    __syncthreads();
    if (j < Cc) out[g * Cc + j] = logits[j] - lse;
}

// ---------------------------------------------------------------- launcher
extern "C" void kernel_launch(void* const* d_in, const int* in_sizes, int n_in,
                              void* d_out, int out_size, void* d_ws, size_t ws_size,
                              hipStream_t stream) {
    (void)in_sizes; (void)n_in; (void)out_size; (void)ws_size;
    const float*     x     = (const float*)d_in[0];
    const long long* eidx  = (const long long*)d_in[1];
    const long long* batch = (const long long*)d_in[2];
    const float*     eps   = (const float*)d_in[3];
    const float*     W1    = (const float*)d_in[4];
    const float*     b1    = (const float*)d_in[5];
    const float*     W2    = (const float*)d_in[6];
    const float*     b2    = (const float*)d_in[7];
    const float*     gamma = (const float*)d_in[8];
    const float*     beta  = (const float*)d_in[9];
    const float*     l1W   = (const float*)d_in[10];
    const float*     l1b   = (const float*)d_in[11];
    const float*     l2W   = (const float*)d_in[12];
    const float*     l2b   = (const float*)d_in[13];

    float* ws       = (float*)d_ws;
    float* h        = ws;                                // N*64
    float* agg      = h + (size_t)Nn * 64;               // N*64
    float* z        = agg + (size_t)Nn * 64;             // N*64
    float* hcat     = z + (size_t)Nn * 64;               // N*320
    float* colsum   = hcat + (size_t)Nn * 320;           // 64
    float* colsumsq = colsum + 64;                       // 64 (contiguous w/ colsum)
    float* mu       = colsumsq + 64;                     // 64
    float* rsig     = mu + 64;                           // 64
    float* pooled   = rsig + 64;                         // G*320
    float* cnt      = pooled + (size_t)Gg * 320;         // G (contiguous w/ pooled)

    copy_f32<<<(Nn * 64 + 255) / 256, 256, 0, stream>>>(x, h, Nn * 64);

    for (int l = 0; l < Ll; ++l) {
        fill_zero<<<(Nn * 64 + 255) / 256, 256, 0, stream>>>(agg, Nn * 64);
        fill_zero<<<1, 256, 0, stream>>>(colsum, 128);  // colsum + colsumsq
        edge_scatter<<<(Ee * 16 + 255) / 256, 256, 0, stream>>>(h, eidx, agg);
        gin_mlp<<<(Nn + 63) / 64, 256, 0, stream>>>(h, agg, eps, l, W1, b1, W2, b2,
                                                    z, colsum, colsumsq);
        bn_stats<<<1, 64, 0, stream>>>(colsum, colsumsq, mu, rsig);
        bn_apply<<<(Nn * 64 + 255) / 256, 256, 0, stream>>>(z, mu, rsig, gamma, beta, l,
                                                            h, hcat);
    }

    fill_zero<<<(Gg * 320 + Gg + 255) / 256, 256, 0, stream>>>(pooled, Gg * 320 + Gg);
    count_nodes<<<(Nn + 255) / 256, 256, 0, stream>>>(batch, cnt);
    pool_sum<<<(Nn * 80 + 255) / 256, 256, 0, stream>>>(hcat, batch, pooled);
    head<<<Gg, 64, 0, stream>>>(pooled, cnt, l1W, l1b, l2W, l2b, (float*)d_out);
}